// SparseInnerProductLayer_55061480735375
// MI455X (gfx1250) — compile-verified
//
#include <hip/hip_runtime.h>
#include <hip/hip_bf16.h>

// SparseInnerProductLayer: out[e] = dot(x[src[e]], x[dst[e]]) for E edges,
// x is [N,128] f32 (51.2 MB -> L2-resident on MI455X's 192 MB L2).
//
// Layout: 8 lanes per edge, 4 edges per wave32 at once.
//   - lane sub=0..7 of group g loads float4 chunks at byte (row*512 + sub*16 + c*128)
//     -> every global_load_b128 covers a contiguous 128B segment per row.
//   - reduction: xor-1/2/4 butterfly inside 8-lane groups via DPP8 register
//     permutes (pure VALU, no LDS, no DS-counter waits).
//   - 4 dots land in lanes {0,8,16,24} at 4 consecutive out addresses -> one
//     masked non-temporal 4-dword store per iteration.
//   - all edge-ID arithmetic in 32-bit; tail clamping compiled only into the
//     (rarely taken) partial-batch path.

#define WAVE   32
#define WPB    8                // waves per block
#define BLOCK  (WAVE * WPB)
#define LPE    8                // lanes per edge
#define EPI    4                // edges per wave iteration
#define ITER   8                // iterations per batch
#define EPB    (EPI * ITER)     // 32 edges per wave batch
#define MAXBLK 2048

#define DPP8_SEL(l0,l1,l2,l3,l4,l5,l6,l7) \
  ((l0)|((l1)<<3)|((l2)<<6)|((l3)<<9)|((l4)<<12)|((l5)<<15)|((l6)<<18)|((l7)<<21))

// Sum across the 8-lane group: xor-1, xor-2, xor-4 butterflies (pure VALU).
__device__ __forceinline__ float group8_reduce_sum(float v) {
#if __has_builtin(__builtin_amdgcn_mov_dpp8)
    v += __int_as_float(__builtin_amdgcn_mov_dpp8(__float_as_int(v),
                                                  DPP8_SEL(1,0,3,2,5,4,7,6)));   // xor 1
    v += __int_as_float(__builtin_amdgcn_mov_dpp8(__float_as_int(v),
                                                  DPP8_SEL(2,3,0,1,6,7,4,5)));   // xor 2
    v += __int_as_float(__builtin_amdgcn_mov_dpp8(__float_as_int(v),
                                                  DPP8_SEL(4,5,6,7,0,1,2,3)));   // xor 4
#else
    v += __shfl_xor(v, 1, WAVE);
    v += __shfl_xor(v, 2, WAVE);
    v += __shfl_xor(v, 4, WAVE);
#endif
    return v;
}

__device__ __forceinline__ float dot4(float4 a, float4 b) {
    float p = a.x * b.x;
    p = fmaf(a.y, b.y, p);
    p = fmaf(a.z, b.z, p);
    p = fmaf(a.w, b.w, p);
    return p;
}

// One 32-edge batch. FULL=true: no clamps, no store bound-checks (hot path).
template <bool FULL>
__device__ __forceinline__ void process_batch(int base, int E, int g, bool sub0,
                                              unsigned sub16,
                                              const char* __restrict__ xb,
                                              const int* __restrict__ srcI,
                                              const int* __restrict__ dstI,
                                              float* __restrict__ out)
{
    // Preload this batch's indices (broadcast within 8-lane groups).
    // All 16 loads touch just 2 cachelines; hoisted for deep pipelining.
    int si[ITER], di[ITER];
#pragma unroll
    for (int t = 0; t < ITER; ++t) {
        int eId = base + t * EPI + g;
        int ec  = FULL ? eId : (eId < E ? eId : E - 1);
        si[t] = srcI[ec];
        di[t] = dstI[ec];
    }

#pragma unroll
    for (int t = 0; t < ITER; ++t) {
        // 32-bit byte offsets against scalar base x (SADDR + voffset form).
        const unsigned voffS = ((unsigned)si[t] << 9) | sub16;
        const unsigned voffD = ((unsigned)di[t] << 9) | sub16;
        const char* ps = xb + voffS;
        const char* pd = xb + voffD;

        const float4 a0 = *(const float4*)(ps      );
        const float4 a1 = *(const float4*)(ps + 128);
        const float4 a2 = *(const float4*)(ps + 256);
        const float4 a3 = *(const float4*)(ps + 384);
        const float4 b0 = *(const float4*)(pd      );
        const float4 b1 = *(const float4*)(pd + 128);
        const float4 b2 = *(const float4*)(pd + 256);
        const float4 b3 = *(const float4*)(pd + 384);

        // 4 independent partial chains for ILP, then combine.
        const float p0 = dot4(a0, b0);
        const float p1 = dot4(a1, b1);
        const float p2 = dot4(a2, b2);
        const float p3 = dot4(a3, b3);
        float p = (p0 + p1) + (p2 + p3);

        p = group8_reduce_sum(p);   // all 8 lanes of the group hold the dot

        // Lanes {0,8,16,24} store 4 consecutive dwords: one 16B NT write.
        const int eId = base + t * EPI + g;
        const bool doStore = FULL ? sub0 : (sub0 && eId < E);
        if (doStore)
            __builtin_nontemporal_store(p, out + eId);
    }
}

__global__ __launch_bounds__(BLOCK)
void sparse_inner_product_kernel(const float* __restrict__ x,
                                 const int* __restrict__ eidx,
                                 float* __restrict__ out,
                                 int E)
{
    const int lane  = threadIdx.x & (WAVE - 1);
    const int sub   = lane & (LPE - 1);          // lane within edge-group (0..7)
    const int g     = lane >> 3;                 // edge-group within wave (0..3)
    const bool sub0 = (sub == 0);
    const unsigned sub16 = (unsigned)sub << 4;   // byte offset of this lane's float4

    const int gwave  = blockIdx.x * WPB + (threadIdx.x >> 5);
    const int stride = gridDim.x * WPB * EPB;

    const int* __restrict__ srcI = eidx;             // edge_index[0][:]
    const int* __restrict__ dstI = eidx + E;         // edge_index[1][:]
    const char* __restrict__ xb  = (const char*)x;   // row stride 512 B

    for (int base = gwave * EPB; base < E; base += stride) {
        if (base + EPB <= E)
            process_batch<true >(base, E, g, sub0, sub16, xb, srcI, dstI, out);
        else
            process_batch<false>(base, E, g, sub0, sub16, xb, srcI, dstI, out);
    }

    // negative_edges placeholder: zeros((1,1)); int32 0 == f32 0.0 bit pattern.
    if (blockIdx.x == 0 && threadIdx.x == 0)
        out[E] = 0.0f;
}

extern "C" void kernel_launch(void* const* d_in, const int* in_sizes, int n_in,
                              void* d_out, int out_size, void* d_ws, size_t ws_size,
                              hipStream_t stream)
{
    (void)n_in; (void)out_size; (void)d_ws; (void)ws_size;

    const float* x    = (const float*)d_in[0];   // [N, 128] f32
    const int*   eidx = (const int*)d_in[1];     // [2, E] int
    float*       out  = (float*)d_out;           // [E] f32 + 1 placeholder

    const int E = in_sizes[1] / 2;

    int batches = (E + EPB - 1) / EPB;
    int blocks  = (batches + WPB - 1) / WPB;
    if (blocks > MAXBLK) blocks = MAXBLK;
    if (blocks < 1)      blocks = 1;

    sparse_inner_product_kernel<<<blocks, BLOCK, 0, stream>>>(x, eidx, out, E);
}